// SPDLogCholeskyMetric_876173328399
// MI455X (gfx1250) — compile-verified
//
#include <hip/hip_runtime.h>
#include <math.h>

// ---------------- configuration ----------------
#define NITER_NS 14           // Newton-Schulz iterations (well-conditioned SPD: eigs >= 1)
#define MATS_PER_BLOCK 16     // one wave per matrix, 512 threads/block
#define MSTRIDE 34            // even padded row stride: b64-aligned A-frags, conflict-free rows
#define MATBUF (MSTRIDE * 32) // 1088 floats
#define YOFF 0
#define ZOFF 1088
#define WOFF 2176             // W area; reused as reduction scratch in the GEMM phase
#define GOFF 1088             // flat-G (1024 floats) reuses the Z area after NS finishes
#define PERMAT 3268           // 3*1088 + 4 pad; PERMAT%64==4 -> distinct banks across matrices

typedef float v2f __attribute__((ext_vector_type(2)));
typedef float v8f __attribute__((ext_vector_type(8)));

// constant-lane broadcast: v_readlane_b32 -> SGPR, no DS traffic
__device__ __forceinline__ float readlane_f(float v, int l) {
    return __builtin_bit_cast(float,
                              __builtin_amdgcn_readlane(__builtin_bit_cast(int, v), l));
}

// ---- WMMA f32 16x16x4 fragment layouts (ISA 7.12.2) ----
// A (16x4): lanes 0-15: M=lane, K={0,1}; lanes 16-31: M=lane-16, K={2,3}
// B (4x16): lanes 0-15: N=lane, K={0,1}; lanes 16-31: N=lane-16, K={2,3}
// C/D (16x16 f32): lanes 0-15: VGPR r = row r; lanes 16-31: VGPR r = row 8+r; col = lane&15

// load the full A-side fragment set of a 32x32 LDS tile: fa[ti][kk]  (b64 loads)
__device__ __forceinline__ void loadFragsA(const float* b, int lo, int hi, v2f fa[2][8]) {
#pragma unroll
    for (int ti = 0; ti < 2; ++ti)
#pragma unroll
        for (int kk = 0; kk < 8; ++kk) {
            int row = ti * 16 + lo, k0 = kk * 4 + 2 * hi;   // even -> 8B aligned
            fa[ti][kk] = *(const v2f*)(b + row * MSTRIDE + k0);
        }
}
// load the full B-side fragment set: fb[kk][tj]  (2addr pairs straight into reg pairs)
__device__ __forceinline__ void loadFragsB(const float* b, int lo, int hi, v2f fb[8][2]) {
#pragma unroll
    for (int kk = 0; kk < 8; ++kk)
#pragma unroll
        for (int tj = 0; tj < 2; ++tj) {
            const float* p = b + (kk * 4 + 2 * hi) * MSTRIDE + tj * 16 + lo;
            v2f r;
            r.x = p[0];
            r.y = p[MSTRIDE];
            fb[kk][tj] = r;
        }
}
// full 32x32 matmul from preloaded fragments: 32 x wmma_f32_16x16x4
__device__ __forceinline__ void mmAll(const v2f fa[2][8], const v2f fb[8][2], v8f acc[4]) {
#pragma unroll
    for (int t = 0; t < 4; ++t) {
        int ti = t >> 1, tj = t & 1;
#pragma unroll
        for (int kk = 0; kk < 8; ++kk)
            acc[t] = __builtin_amdgcn_wmma_f32_16x16x4_f32(false, fa[ti][kk], false,
                                                           fb[kk][tj], (short)0, acc[t],
                                                           false, false);
    }
}

// MODE 0: write G = logchol(sqrtm(S)) flat [nmat][1024] to outp (used for P)
// MODE 1: fused: G for 16 matrices, then GEMM vs Bmat(A) -> logits [nmat][128]
template <int MODE>
__global__ __launch_bounds__(512) void spd_g_kernel(
    const float* __restrict__ Sglob,  // [nmat][32][32] SPD, row-major
    const float* __restrict__ Aglob,  // [128][32][32] symmetric (MODE 1)
    const float* __restrict__ tp,     // [128] term_P (MODE 1)
    float* __restrict__ outp,         // MODE0: [nmat][1024] ; MODE1: [nmat][128]
    int nmat) {
    extern __shared__ float ls[];
    const int wave = threadIdx.x >> 5;
    const int lane = threadIdx.x & 31;
    const int lo = lane & 15, hi = lane >> 4;
    const int gm = blockIdx.x * MATS_PER_BLOCK + wave;
    float* Yb = ls + wave * PERMAT + YOFF;
    float* Zb = ls + wave * PERMAT + ZOFF;
    float* Wb = ls + wave * PERMAT + WOFF;
    float* Gf = ls + wave * PERMAT + GOFF;

    if (gm < nmat) {  // wave-uniform branch: EXEC stays all-ones for WMMA
        // ---- load row `lane` of S (b128s); Frobenius norm for NS scaling ----
        const float4* S4 = (const float4*)(Sglob + (size_t)gm * 1024 + lane * 32);
        float r[32];
        float ss = 0.f;
#pragma unroll
        for (int q = 0; q < 8; ++q) {
            float4 v = S4[q];
            r[4 * q + 0] = v.x; r[4 * q + 1] = v.y;
            r[4 * q + 2] = v.z; r[4 * q + 3] = v.w;
            ss += v.x * v.x + v.y * v.y + v.z * v.z + v.w * v.w;
        }
#pragma unroll
        for (int off = 16; off > 0; off >>= 1) ss += __shfl_xor(ss, off, 32);
        float s  = sqrtf(ss);      // ||S||_F : eigs of S/s in (0,1]
        float rs = 1.f / s;
        float sq = sqrtf(s);
#pragma unroll
        for (int q = 0; q < 16; ++q) {   // b64 seed stores (row base lane*34 is even)
            v2f y; y.x = r[2 * q] * rs; y.y = r[2 * q + 1] * rs;
            *(v2f*)(Yb + lane * MSTRIDE + 2 * q) = y;
            v2f z; z.x = (lane == 2 * q) ? 1.f : 0.f; z.y = (lane == 2 * q + 1) ? 1.f : 0.f;
            *(v2f*)(Zb + lane * MSTRIDE + 2 * q) = z;
        }
        // ---- Newton-Schulz coupled iteration: Y -> sqrtm(S/s) ----
        for (int it = 0; it < NITER_NS; ++it) {
            v2f fa[2][8], fb[8][2];
            // W = 3I - Z*Y
            loadFragsA(Zb, lo, hi, fa);
            loadFragsB(Yb, lo, hi, fb);
            {
                v8f w[4] = {};
                mmAll(fa, fb, w);
#pragma unroll
                for (int t = 0; t < 4; ++t) {
                    int ti = t >> 1, tj = t & 1;
                    float* p = Wb + (ti * 16 + 8 * hi) * MSTRIDE + tj * 16 + lo;
                    int row0 = ti * 16 + 8 * hi, col = tj * 16 + lo;
#pragma unroll
                    for (int rr = 0; rr < 8; ++rr)
                        p[rr * MSTRIDE] = ((row0 + rr == col) ? 3.f : 0.f) - w[t][rr];
                }
            }
            // Y' = 0.5 * Y * W
            loadFragsA(Yb, lo, hi, fa);
            loadFragsB(Wb, lo, hi, fb);
            {
                v8f ay[4] = {};
                mmAll(fa, fb, ay);
#pragma unroll
                for (int t = 0; t < 4; ++t) {
                    int ti = t >> 1, tj = t & 1;
                    float* p = Yb + (ti * 16 + 8 * hi) * MSTRIDE + tj * 16 + lo;
#pragma unroll
                    for (int rr = 0; rr < 8; ++rr) p[rr * MSTRIDE] = 0.5f * ay[t][rr];
                }
            }
            // Z' = 0.5 * W * Z
            loadFragsA(Wb, lo, hi, fa);
            loadFragsB(Zb, lo, hi, fb);
            {
                v8f az[4] = {};
                mmAll(fa, fb, az);
#pragma unroll
                for (int t = 0; t < 4; ++t) {
                    int ti = t >> 1, tj = t & 1;
                    float* p = Zb + (ti * 16 + 8 * hi) * MSTRIDE + tj * 16 + lo;
#pragma unroll
                    for (int rr = 0; rr < 8; ++rr) p[rr * MSTRIDE] = 0.5f * az[t][rr];
                }
            }
        }

        // ---- register-resident Cholesky of M = sqrt(s)*Y, lane = row ----
        // Fully unrolled: row lives in 32 VGPRs with constant indices; all
        // cross-lane traffic is v_readlane_b32 (SGPR broadcast) -> zero DS ops.
        float row[32];
#pragma unroll
        for (int q = 0; q < 16; ++q) {
            v2f y = *(const v2f*)(Yb + lane * MSTRIDE + 2 * q);   // b64 loads
            row[2 * q]     = y.x * sq;
            row[2 * q + 1] = y.y * sq;
        }
        float mydiag = 1.f;
#pragma unroll
        for (int j = 0; j < 32; ++j) {
            float mjj = readlane_f(row[j], j);
            float djj = sqrtf(mjj);
            float inv = 1.0f / djj;
            float lij = (lane > j) ? row[j] * inv : 0.f;
            row[j]  = (lane == j) ? djj : lij;     // lanes < j get 0 (masked below anyway)
            mydiag  = (lane == j) ? djj : mydiag;  // defer logf to after the loop
#pragma unroll
            for (int c = j + 1; c < 32; ++c) {
                float lcj = readlane_f(row[j], c);  // L[c][j] via readlane
                row[c] -= lij * lcj;                // harmless for lanes < c (masked below)
            }
        }
        float dlog = logf(mydiag);                  // single logf per lane
        // ---- log-Cholesky map -> flat G[1024] via 8 x ds_store_b128 ----
#pragma unroll
        for (int q = 0; q < 8; ++q) {
            float4 w;
            { int j = 4 * q + 0; w.x = (j < lane) ? row[j] : (j == lane ? dlog : 0.f); }
            { int j = 4 * q + 1; w.y = (j < lane) ? row[j] : (j == lane ? dlog : 0.f); }
            { int j = 4 * q + 2; w.z = (j < lane) ? row[j] : (j == lane ? dlog : 0.f); }
            { int j = 4 * q + 3; w.w = (j < lane) ? row[j] : (j == lane ? dlog : 0.f); }
            *(float4*)(Gf + lane * 32 + 4 * q) = w;   // 16B aligned
        }
    } else {
        // pad rows so the GEMM A-tile reads zeros
        float4 zz = {0.f, 0.f, 0.f, 0.f};
#pragma unroll
        for (int q = 0; q < 8; ++q) *(float4*)(Gf + lane * 32 + 4 * q) = zz;
    }

    if (MODE == 0) {
        if (gm < nmat) {
            float4* dst = (float4*)(outp + (size_t)gm * 1024 + lane * 32);
#pragma unroll
            for (int q = 0; q < 8; ++q) dst[q] = *(const float4*)(Gf + lane * 32 + 4 * q);
        }
        return;
    }

    // ---- fused logits GEMM: [16 x 1024] G-tile  x  [1024 x 128] Bmat^T ----
    // 16 waves: wave = (half<<3)|tj ; pair (tj, tj+8) splits K, reduces through LDS.
    __syncthreads();
    const int tj = wave & 7, half = wave >> 3;
    const int cls = tj * 16 + lo;
    v8f acc = {};
    for (int kk = half * 128; kk < half * 128 + 128; ++kk) {
        int k0 = kk * 4 + 2 * hi;  // even element pair index
        // A fragment: row = matrix index (lo), K = {k0, k0+1}  (b64 LDS load)
        v2f a = *(const v2f*)(ls + lo * PERMAT + GOFF + k0);
        // B fragment: Bmat[cls][k0..k0+1] = tril_half_diag(A) built on the fly
        int i = k0 >> 5, j = k0 & 31;  // j even, both elems in row i
        const float2 va = *(const float2*)(Aglob + (size_t)cls * 1024 + k0);
        v2f b;
        b.x = (i > j)     ? va.x : ((i == j)     ? 0.5f * va.x : 0.f);
        b.y = (i > j + 1) ? va.y : ((i == j + 1) ? 0.5f * va.y : 0.f);
        acc = __builtin_amdgcn_wmma_f32_16x16x4_f32(false, a, false, b, (short)0, acc,
                                                    false, false);
    }
    if (half == 1) {  // stash partial in own (free) W area
        float* red = ls + wave * PERMAT + WOFF;
#pragma unroll
        for (int rr = 0; rr < 8; ++rr) red[lane * 8 + rr] = acc[rr];
    }
    __syncthreads();
    if (half == 0) {
        const float* red = ls + (wave + 8) * PERMAT + WOFF;
#pragma unroll
        for (int rr = 0; rr < 8; ++rr) acc[rr] += red[lane * 8 + rr];
        float tpk = tp[cls];
#pragma unroll
        for (int rr = 0; rr < 8; ++rr) {
            int m = 8 * hi + rr;  // matrix row within block
            size_t gmo = (size_t)blockIdx.x * MATS_PER_BLOCK + m;
            if (gmo < (size_t)nmat)
                outp[gmo * 128 + cls] = (acc[rr] - tpk) * 2.0f;  // /POWER, POWER=0.5
        }
    }
}

// term_P[k] = <G_P[k], Bmat[k]>  (tiny: 128 x 1024 dot)
__global__ void termP_kernel(const float* __restrict__ Gp, const float* __restrict__ Aglob,
                             float* __restrict__ tp) {
    int k = threadIdx.x;  // 128 threads
    float s = 0.f;
    for (int e = 0; e < 1024; ++e) {
        int i = e >> 5, j = e & 31;
        float av = Aglob[k * 1024 + e];
        float bv = (i > j) ? av : ((i == j) ? 0.5f * av : 0.f);
        s += Gp[k * 1024 + e] * bv;
    }
    tp[k] = s;
}

extern "C" void kernel_launch(void* const* d_in, const int* in_sizes, int n_in,
                              void* d_out, int out_size, void* d_ws, size_t ws_size,
                              hipStream_t stream) {
    const float* S = (const float*)d_in[0];  // [8192,4,32,32]
    const float* P = (const float*)d_in[1];  // [128,32,32]
    const float* A = (const float*)d_in[2];  // [128,32,32]
    float* out = (float*)d_out;              // [8192,4,128]
    float* Gp = (float*)d_ws;                // 128*1024 floats  (512 KB)
    float* tp = Gp + 128 * 1024;             // 128 floats

    const int NMAT_P = 128;
    const int NMAT_S = 8192 * 4;
    const size_t lds = (size_t)MATS_PER_BLOCK * PERMAT * sizeof(float);  // ~204 KB / 320 KB WGP

    // opt in to >64KB dynamic LDS (CDNA5 WGP has 320KB)
    hipFuncSetAttribute(reinterpret_cast<const void*>(spd_g_kernel<0>),
                        hipFuncAttributeMaxDynamicSharedMemorySize, (int)lds);
    hipFuncSetAttribute(reinterpret_cast<const void*>(spd_g_kernel<1>),
                        hipFuncAttributeMaxDynamicSharedMemorySize, (int)lds);

    // 1) G_P for the 128 class prototypes
    spd_g_kernel<0><<<NMAT_P / MATS_PER_BLOCK, 512, lds, stream>>>(P, nullptr, nullptr, Gp,
                                                                   NMAT_P);
    // 2) term_P scalars
    termP_kernel<<<1, 128, 0, stream>>>(Gp, A, tp);
    // 3) fused sqrtm + log-Cholesky + logits GEMM over all 32768 S matrices
    spd_g_kernel<1><<<NMAT_S / MATS_PER_BLOCK, 512, lds, stream>>>(S, A, tp, out, NMAT_S);
}